// PointNetClassificationMSG_31413390803132
// MI455X (gfx1250) — compile-verified
//
#include <hip/hip_runtime.h>
#include <hip/hip_bf16.h>
#include <math.h>

typedef float v2f __attribute__((ext_vector_type(2)));
typedef float v8f __attribute__((ext_vector_type(8)));

#define BN_EPS 1e-5f

// ---------------------------------------------------------------------------
// Farthest point sampling: one block per batch. dists kept in LDS, argmax via
// tree reduction with lowest-index tie-break (matches jnp.argmax).
// ---------------------------------------------------------------------------
__global__ void fps_kernel(const float* __restrict__ xyz, int* __restrict__ out,
                           int N, int npoint) {
  __shared__ float sdist[4096];
  __shared__ float rv[256];
  __shared__ int   ri[256];
  const int b = blockIdx.x;
  const int t = threadIdx.x;
  const float* P = xyz + (size_t)b * N * 3;
  for (int j = t; j < N; j += 256) sdist[j] = 1e10f;
  __syncthreads();
  int far = 0;
  for (int i = 0; i < npoint; ++i) {
    if (t == 0) out[b * npoint + i] = far;
    const float px = P[far * 3 + 0];
    const float py = P[far * 3 + 1];
    const float pz = P[far * 3 + 2];
    float bd = -1.0f; int bj = 0;
    for (int j = t; j < N; j += 256) {
      const float dx = P[j * 3 + 0] - px;
      const float dy = P[j * 3 + 1] - py;
      const float dz = P[j * 3 + 2] - pz;
      const float d  = dx * dx + dy * dy + dz * dz;
      const float nd = fminf(sdist[j], d);
      sdist[j] = nd;
      if (nd > bd) { bd = nd; bj = j; }   // strict > keeps lowest index per thread
    }
    rv[t] = bd; ri[t] = bj;
    __syncthreads();
    for (int s = 128; s > 0; s >>= 1) {
      if (t < s) {
        if (rv[t + s] > rv[t] || (rv[t + s] == rv[t] && ri[t + s] < ri[t])) {
          rv[t] = rv[t + s]; ri[t] = ri[t + s];
        }
      }
      __syncthreads();
    }
    far = ri[0];
    __syncthreads();
  }
}

// nx[b,s,c] = xyz[b, fi[b,s], c]
__global__ void gather_centers_kernel(const float* __restrict__ xyz,
                                      const int* __restrict__ fi,
                                      float* __restrict__ nx, int N, int S, int B) {
  const int i = blockIdx.x * blockDim.x + threadIdx.x;
  if (i >= B * S * 3) return;
  const int c  = i % 3;
  const int bs = i / 3;
  const int s  = bs % S;
  const int b  = bs / S;
  nx[i] = xyz[((size_t)b * N + fi[b * S + s]) * 3 + c];
}

// ---------------------------------------------------------------------------
// Ball query: one wave32 per center. Appends the first `ns` indices (in index
// order) with d2 < r2, then duplicates the first hit into unused slots.
// ---------------------------------------------------------------------------
__global__ void ball_query_kernel(const float* __restrict__ xyz,
                                  const float* __restrict__ nx,
                                  int* __restrict__ idx,
                                  int B, int N, int S, int ns, float r2) {
  const int lane = threadIdx.x & 31;
  const int wid  = (blockIdx.x * blockDim.x + threadIdx.x) >> 5;
  if (wid >= B * S) return;
  const int b = wid / S;
  const int s = wid % S;
  const float* P = xyz + (size_t)b * N * 3;
  const float cx = nx[(b * S + s) * 3 + 0];
  const float cy = nx[(b * S + s) * 3 + 1];
  const float cz = nx[(b * S + s) * 3 + 2];
  int* o = idx + (size_t)(b * S + s) * ns;
  int cnt = 0, firstIdx = 0;
  for (int base = 0; base < N && cnt < ns; base += 32) {
    const int i = base + lane;
    bool hit = false;
    if (i < N) {
      const float dx = P[i * 3 + 0] - cx;
      const float dy = P[i * 3 + 1] - cy;
      const float dz = P[i * 3 + 2] - cz;
      hit = (dx * dx + dy * dy + dz * dz) < r2;
    }
    const unsigned mask = (unsigned)__ballot(hit);
    if (cnt == 0 && mask) firstIdx = base + __ffs(mask) - 1;
    const int pos = cnt + __popc(mask & ((1u << lane) - 1u));
    if (hit && pos < ns) o[pos] = i;
    cnt += __popc(mask);
  }
  if (cnt < ns) {
    for (int j = cnt + lane; j < ns; j += 32) o[j] = firstIdx;
  }
}

// Grouped features with padded leading dim lda (zero-filled pad columns):
// g[(s*K+k)*lda + c] = c<3 ? xyz[idx]-center : (c<3+Cf ? feat[idx,c-3] : 0)
__global__ void group_kernel(const float* __restrict__ xyzB,
                             const float* __restrict__ featB, int Cf,
                             const float* __restrict__ nxB,
                             const int* __restrict__ idxB,
                             float* __restrict__ g, int S, int K, int lda) {
  const int C = 3 + Cf;
  const int total = S * K * lda;
  for (int i = blockIdx.x * blockDim.x + threadIdx.x; i < total;
       i += gridDim.x * blockDim.x) {
    const int c  = i % lda;
    const int sk = i / lda;
    const int k  = sk % K;
    const int s  = sk / K;
    float v = 0.0f;
    if (c < C) {
      const int j = idxB[s * K + k];
      if (c < 3) v = xyzB[j * 3 + c] - nxB[s * 3 + c];
      else       v = featB[(size_t)j * Cf + (c - 3)];
    }
    g[i] = v;
  }
}

// sa3 "group all" with padded lda: g[k*lda+c] = c<3 ? nx[k,c] : feat[k,c-3]
__global__ void concat_kernel(const float* __restrict__ nxB,
                              const float* __restrict__ fB,
                              float* __restrict__ g, int K, int Cf, int lda) {
  const int C = 3 + Cf;
  const int total = K * lda;
  const int i = blockIdx.x * blockDim.x + threadIdx.x;
  if (i >= total) return;
  const int c = i % lda;
  const int k = i / lda;
  float v = 0.0f;
  if (c < 3)      v = nxB[k * 3 + c];
  else if (c < C) v = fB[(size_t)k * Cf + (c - 3)];
  g[i] = v;
}

// Pack weight (K,N) row-major into zero-padded WMMA B-fragment order:
//   Wf[(((k/4)*2 + (k%4)/2) * Np + n) * 2 + (k%2)] = W[k][n]
// so a lane's two K-values for one fragment are one contiguous 8B load.
__global__ void pack_weight_kernel(const float* __restrict__ W, float* __restrict__ Wf,
                                   int K, int N, int Kp, int Np) {
  const int i = blockIdx.x * blockDim.x + threadIdx.x;
  if (i >= Kp * Np) return;
  const int j   = i & 1;                 // k LSB
  const int n   = (i >> 1) % Np;         // column
  const int h   = ((i >> 1) / Np) & 1;   // lane-half (kh/2)
  const int k0g = (i >> 1) / Np / 2;     // k-group (k0/4)
  const int k   = k0g * 4 + h * 2 + j;
  Wf[i] = (k < K && n < N) ? W[(size_t)k * N + n] : 0.0f;
}

// ---------------------------------------------------------------------------
// WMMA fp32 GEMM: out(M,N) = A(M,*) * Wf(frag-packed Kp x Np), fused epilogue.
// A has leading dim lda (mult of 4, zero-padded K columns); Wf is zero-padded
// and pre-packed in B-fragment order. Each wave computes GEMM_MT 16x16 M-tiles
// sharing one B fragment: steady state is 1x b64 (B) + 4x b64 (A) -> 4x v_wmma.
// Row OOB handled by clamping (loads valid, stores guarded).
//   mode 0: relu((x - m) * rsqrt(v+eps) * g + b)
//   mode 1: x + bias
// ---------------------------------------------------------------------------
#define GEMM_WAVES 4
#define GEMM_MT 4
__global__ void gemm_kernel(const float* __restrict__ A, int lda,
                            const float* __restrict__ Wf, int Np,
                            const float* __restrict__ bng, const float* __restrict__ bnb,
                            const float* __restrict__ bnm, const float* __restrict__ bnv,
                            const float* __restrict__ bias,
                            float* __restrict__ out,
                            int M, int N, int Kp, int mode) {
  const int lane  = threadIdx.x & 31;
  const int wave  = threadIdx.x >> 5;
  const int mbase = (blockIdx.x * GEMM_WAVES + wave) * (GEMM_MT * 16);
  const int ntile = blockIdx.y;
  if (mbase >= M) return;                     // wave-uniform: EXEC stays all-1
  const int col = ntile * 16 + (lane & 15);   // always < Np (padded)
  const int kh  = (lane >> 4) * 2;            // 0 or 2

  const float* Ap[GEMM_MT];
#pragma unroll
  for (int i = 0; i < GEMM_MT; ++i) {
    int r = mbase + i * 16 + (lane & 15);
    if (r > M - 1) r = M - 1;                 // clamp: keep loads in-bounds
    Ap[i] = A + (size_t)r * lda + kh;
  }
  // B fragment base for this lane: ((lane>>4)*Np + col)*2; stride 4*Np per k-group
  const float* Bp = Wf + ((size_t)(lane >> 4) * Np + col) * 2;

  v8f acc[GEMM_MT];
#pragma unroll
  for (int i = 0; i < GEMM_MT; ++i) acc[i] = (v8f){0.f,0.f,0.f,0.f,0.f,0.f,0.f,0.f};

  const int kgroups = Kp >> 2;
  for (int kg = 0; kg < kgroups; ++kg) {
    const v2f b = *(const v2f*)(Bp + (size_t)kg * 4 * Np);
    v2f a[GEMM_MT];
#pragma unroll
    for (int i = 0; i < GEMM_MT; ++i) a[i] = *(const v2f*)(Ap[i] + (size_t)kg * 4);
#pragma unroll
    for (int i = 0; i < GEMM_MT; ++i) {
      acc[i] = __builtin_amdgcn_wmma_f32_16x16x4_f32(false, a[i], false, b,
                                                     (short)0, acc[i], false, false);
    }
  }

  float s = 1.0f, t = 0.0f;
  if (col < N) {
    if (mode == 0) { s = bng[col] * rsqrtf(bnv[col] + BN_EPS); t = bnb[col] - bnm[col] * s; }
    else           { t = bias[col]; }
  }
#pragma unroll
  for (int i = 0; i < GEMM_MT; ++i) {
    const int mb = mbase + i * 16 + ((lane >> 4) << 3);
#pragma unroll
    for (int r = 0; r < 8; ++r) {
      const int m = mb + r;
      if (m < M && col < N) {
        float vv = acc[i][r] * s + t;
        if (mode == 0) vv = fmaxf(vv, 0.0f);
        out[(size_t)m * N + col] = vv;
      }
    }
  }
}

// out[s*Ctot + coff + c] = max_k act[(s*K+k)*C + c]     (per batch)
__global__ void maxpool_kernel(const float* __restrict__ act, float* __restrict__ out,
                               int S, int K, int C, int coff, int Ctot) {
  const int i = blockIdx.x * blockDim.x + threadIdx.x;
  if (i >= S * C) return;
  const int s = i / C;
  const int c = i % C;
  const float* p = act + (size_t)s * K * C + c;
  float m = p[0];
  for (int k = 1; k < K; ++k) m = fmaxf(m, p[(size_t)k * C]);
  out[(size_t)s * Ctot + coff + c] = m;
}

// ---------------------------------------------------------------------------
struct BnW { const float *g, *b, *m, *v, *W; };
struct Lay { const float *g, *b, *m, *v; const float* Wf; int Kp, Np; };

static inline void run_gemm(const float* A, int lda, const Lay& L, const float* bias,
                            float* out, int M, int N, int mode, hipStream_t st) {
  const int Mtiles = (M + 15) / 16;
  const int span   = GEMM_WAVES * GEMM_MT;            // m-tiles per block
  dim3 grid((Mtiles + span - 1) / span, L.Np / 16);
  gemm_kernel<<<grid, GEMM_WAVES * 32, 0, st>>>(A, lda, L.Wf, L.Np,
                                                L.g, L.b, L.m, L.v, bias,
                                                out, M, N, L.Kp, mode);
}

extern "C" void kernel_launch(void* const* d_in, const int* in_sizes, int n_in,
                              void* d_out, int out_size, void* d_ws, size_t ws_size,
                              hipStream_t stream) {
  (void)in_sizes; (void)n_in; (void)out_size; (void)ws_size;
  // ---- input mapping: setup_inputs() insertion order --------------------
  int p = 0;
  const float* xyz = (const float*)d_in[p++];            // (8,4096,3)
  BnW sa1[3][3], sa2[3][3], sa3r[3];
  auto take = [&](BnW& L) {
    L.g = (const float*)d_in[p + 0];
    L.b = (const float*)d_in[p + 1];
    L.m = (const float*)d_in[p + 2];
    L.v = (const float*)d_in[p + 3];
    L.W = (const float*)d_in[p + 4];
    p += 5;
  };
  for (int i = 0; i < 3; ++i) for (int j = 0; j < 3; ++j) take(sa1[i][j]);
  for (int i = 0; i < 3; ++i) for (int j = 0; j < 3; ++j) take(sa2[i][j]);
  for (int j = 0; j < 3; ++j) take(sa3r[j]);
  BnW fc1, fc2, fc3;
  fc1.W = (const float*)d_in[p++];
  fc1.g = (const float*)d_in[p++]; fc1.b = (const float*)d_in[p++];
  fc1.m = (const float*)d_in[p++]; fc1.v = (const float*)d_in[p++];
  fc2.W = (const float*)d_in[p++];
  fc2.g = (const float*)d_in[p++]; fc2.b = (const float*)d_in[p++];
  fc2.m = (const float*)d_in[p++]; fc2.v = (const float*)d_in[p++];
  fc3.W = (const float*)d_in[p++];
  fc3.g = fc3.b = fc3.m = fc3.v = nullptr;
  const float* b3 = (const float*)d_in[p++];

  // ---- workspace carve ---------------------------------------------------
  char* wsb = (char*)d_ws; size_t off = 0;
  auto carve = [&](size_t bytes) -> void* {
    void* r = wsb + off; off = (off + bytes + 255) & ~((size_t)255); return r;
  };
  int*   fi1  = (int*)  carve((size_t)8 * 512 * sizeof(int));
  float* nx1  = (float*)carve((size_t)8 * 512 * 3 * sizeof(float));
  int*   fi2  = (int*)  carve((size_t)8 * 128 * sizeof(int));
  float* nx2  = (float*)carve((size_t)8 * 128 * 3 * sizeof(float));
  int*   idxb = (int*)  carve((size_t)8 * 512 * 128 * sizeof(int));
  float* f1   = (float*)carve((size_t)8 * 512 * 320 * sizeof(float));
  float* f2   = (float*)carve((size_t)8 * 128 * 640 * sizeof(float));
  float* f3   = (float*)carve((size_t)8 * 1024 * sizeof(float));
  float* fcA  = (float*)carve((size_t)8 * 512 * sizeof(float));
  float* fcB  = (float*)carve((size_t)8 * 256 * sizeof(float));
  float* bufG = (float*)carve((size_t)5308416 * sizeof(float)); // max 128*128*324
  float* bufA = (float*)carve((size_t)8388608 * sizeof(float)); // max 512*128*128
  float* bufB = (float*)carve((size_t)6291456 * sizeof(float)); // max 512*128*96

  auto ceil4  = [](int x) { return (x + 3)  & ~3;  };
  auto ceil16 = [](int x) { return (x + 15) & ~15; };
  auto prep = [&](const BnW& src, int K, int N) -> Lay {
    Lay L; L.g = src.g; L.b = src.b; L.m = src.m; L.v = src.v;
    L.Kp = ceil4(K); L.Np = ceil16(N);
    float* wf = (float*)carve((size_t)L.Kp * L.Np * sizeof(float));
    pack_weight_kernel<<<((size_t)L.Kp * L.Np + 255) / 256, 256, 0, stream>>>(
        src.W, wf, K, N, L.Kp, L.Np);
    L.Wf = wf;
    return L;
  };

  const int sa1_c[3][4] = {{3, 32, 32, 64}, {3, 64, 64, 128}, {3, 64, 96, 128}};
  const int sa2_c[3][4] = {{323, 64, 64, 128}, {323, 128, 128, 256}, {323, 128, 128, 256}};
  const int sa3_c[4]    = {643, 256, 512, 1024};
  Lay L1[3][3], L2[3][3], L3[3], F1, F2, F3;
  for (int sc = 0; sc < 3; ++sc)
    for (int l = 0; l < 3; ++l) L1[sc][l] = prep(sa1[sc][l], sa1_c[sc][l], sa1_c[sc][l + 1]);
  for (int sc = 0; sc < 3; ++sc)
    for (int l = 0; l < 3; ++l) L2[sc][l] = prep(sa2[sc][l], sa2_c[sc][l], sa2_c[sc][l + 1]);
  for (int l = 0; l < 3; ++l) L3[l] = prep(sa3r[l], sa3_c[l], sa3_c[l + 1]);
  F1 = prep(fc1, 1024, 512);
  F2 = prep(fc2, 512, 256);
  F3 = prep(fc3, 256, 40);

  // ======================= SA1 (N=4096 -> S=512) =========================
  fps_kernel<<<8, 256, 0, stream>>>(xyz, fi1, 4096, 512);
  gather_centers_kernel<<<(8 * 512 * 3 + 255) / 256, 256, 0, stream>>>(
      xyz, fi1, nx1, 4096, 512, 8);
  {
    const int   ns_t[3] = {16, 32, 128};
    const float r_t[3]  = {0.1f, 0.2f, 0.4f};
    const int   coff[3] = {0, 64, 192};
    for (int sc = 0; sc < 3; ++sc) {
      const int ns = ns_t[sc]; const float r = r_t[sc];
      ball_query_kernel<<<(8 * 512 + 7) / 8, 256, 0, stream>>>(
          xyz, nx1, idxb, 8, 4096, 512, ns, r * r);
      for (int b = 0; b < 8; ++b) {
        const float* xb = xyz  + (size_t)b * 4096 * 3;
        const float* nb = nx1  + (size_t)b * 512 * 3;
        const int*   ib = idxb + (size_t)b * 512 * ns;
        const int SK = 512 * ns;
        const int lda0 = 4; // ceil4(3)
        group_kernel<<<((size_t)SK * lda0 + 255) / 256, 256, 0, stream>>>(
            xb, nullptr, 0, nb, ib, bufG, 512, ns, lda0);
        run_gemm(bufG, lda0,          L1[sc][0], nullptr, bufA, SK, sa1_c[sc][1], 0, stream);
        run_gemm(bufA, sa1_c[sc][1],  L1[sc][1], nullptr, bufB, SK, sa1_c[sc][2], 0, stream);
        run_gemm(bufB, sa1_c[sc][2],  L1[sc][2], nullptr, bufA, SK, sa1_c[sc][3], 0, stream);
        maxpool_kernel<<<(512 * sa1_c[sc][3] + 255) / 256, 256, 0, stream>>>(
            bufA, f1 + (size_t)b * 512 * 320, 512, ns, sa1_c[sc][3], coff[sc], 320);
      }
    }
  }

  // ======================= SA2 (N=512 -> S=128) ==========================
  fps_kernel<<<8, 256, 0, stream>>>(nx1, fi2, 512, 128);
  gather_centers_kernel<<<(8 * 128 * 3 + 255) / 256, 256, 0, stream>>>(
      nx1, fi2, nx2, 512, 128, 8);
  {
    const int   ns_t[3] = {32, 64, 128};
    const float r_t[3]  = {0.2f, 0.4f, 0.8f};
    const int   coff[3] = {0, 128, 384};
    for (int sc = 0; sc < 3; ++sc) {
      const int ns = ns_t[sc]; const float r = r_t[sc];
      ball_query_kernel<<<(8 * 128 + 7) / 8, 256, 0, stream>>>(
          nx1, nx2, idxb, 8, 512, 128, ns, r * r);
      for (int b = 0; b < 8; ++b) {
        const float* xb = nx1  + (size_t)b * 512 * 3;
        const float* fb = f1   + (size_t)b * 512 * 320;
        const float* nb = nx2  + (size_t)b * 128 * 3;
        const int*   ib = idxb + (size_t)b * 128 * ns;
        const int SK = 128 * ns;
        const int lda0 = 324; // ceil4(323)
        group_kernel<<<((size_t)SK * lda0 + 255) / 256, 256, 0, stream>>>(
            xb, fb, 320, nb, ib, bufG, 128, ns, lda0);
        run_gemm(bufG, lda0,          L2[sc][0], nullptr, bufA, SK, sa2_c[sc][1], 0, stream);
        run_gemm(bufA, sa2_c[sc][1],  L2[sc][1], nullptr, bufB, SK, sa2_c[sc][2], 0, stream);
        run_gemm(bufB, sa2_c[sc][2],  L2[sc][2], nullptr, bufA, SK, sa2_c[sc][3], 0, stream);
        maxpool_kernel<<<(128 * sa2_c[sc][3] + 255) / 256, 256, 0, stream>>>(
            bufA, f2 + (size_t)b * 128 * 640, 128, ns, sa2_c[sc][3], coff[sc], 640);
      }
    }
  }

  // ======================= SA3 (group-all, K=128) ========================
  for (int b = 0; b < 8; ++b) {
    const int lda0 = 644; // ceil4(643)
    concat_kernel<<<(128 * lda0 + 255) / 256, 256, 0, stream>>>(
        nx2 + (size_t)b * 128 * 3, f2 + (size_t)b * 128 * 640, bufG, 128, 640, lda0);
    run_gemm(bufG, lda0, L3[0], nullptr, bufA, 128, 256,  0, stream);
    run_gemm(bufA, 256,  L3[1], nullptr, bufB, 128, 512,  0, stream);
    run_gemm(bufB, 512,  L3[2], nullptr, bufA, 128, 1024, 0, stream);
    maxpool_kernel<<<(1024 + 255) / 256, 256, 0, stream>>>(
        bufA, f3 + (size_t)b * 1024, 1, 128, 1024, 0, 1024);
  }

  // ======================= FC head =======================================
  run_gemm(f3,  1024, F1, nullptr, fcA, 8, 512, 0, stream);
  run_gemm(fcA, 512,  F2, nullptr, fcB, 8, 256, 0, stream);
  run_gemm(fcB, 256,  F3, b3, (float*)d_out, 8, 40, 1, stream);
}